// SingleDimHistLayer_36172214567794
// MI455X (gfx1250) — compile-verified
//
#include <hip/hip_runtime.h>

// Soft histogram, telescoped form:
//   p[b,k] = 0.5 * (T[b,k] - T[b,k+1]) / N,
//   T[b,k] = sum_pixels tanh((x - k/255) * INV2W),  INV2W = 255*2.5/2
// One v_tanh_f32 per (pixel, edge): 268M trans ops total (4x fewer than naive).

typedef float v2f __attribute__((ext_vector_type(2)));
typedef float v8f __attribute__((ext_vector_type(8)));

#define INV2W 318.75f  // 1/(2W) = K/(2*0.4/K) ... = 255*2.5/2

__device__ __forceinline__ float htanh(float x) {
#if __has_builtin(__builtin_amdgcn_tanhf)
  return __builtin_amdgcn_tanhf(x);
#elif __has_builtin(__builtin_amdgcn_tanh_f32)
  return __builtin_amdgcn_tanh_f32(x);
#else
  float r;
  asm volatile("v_tanh_f32 %0, %1\n\tv_nop" : "=v"(r) : "v"(x));
  return r;
#endif
}

__global__ __launch_bounds__(256) void hist_accum(const float* __restrict__ x,
                                                  float* __restrict__ T,
                                                  int N, int pixPerBlock) {
  __shared__ float lds[256 * 8];  // [edge][wave]
  const int tid   = threadIdx.x;
  const int lane  = tid & 31;
  const int wave  = __builtin_amdgcn_readfirstlane(tid >> 5);  // wave-uniform SGPR
  const int batch = blockIdx.y;
  const int pixPerWave = pixPerBlock >> 3;  // 8 waves per block

  float acc[8];
  float c[8];
#pragma unroll
  for (int j = 0; j < 8; ++j) {
    acc[j] = 0.0f;
    // edge_k * INV2W with k = lane + 32*j:  (k/255)*318.75 = k*1.25
    c[j] = (float)(lane + 32 * j) * 1.25f;
  }

  // Wave-uniform pixel pointer -> scalar (s_load) broadcast loads.
  const float* xb = x + (size_t)batch * (size_t)N
                      + (size_t)blockIdx.x * (size_t)pixPerBlock
                      + (size_t)wave * (size_t)pixPerWave;

  for (int i = 0; i < pixPerWave; i += 4) {
    const float4 px = *(const float4*)(xb + i);
    const float xs0 = px.x * INV2W;
    const float xs1 = px.y * INV2W;
    const float xs2 = px.z * INV2W;
    const float xs3 = px.w * INV2W;
#pragma unroll
    for (int j = 0; j < 8; ++j) {
      acc[j] += htanh(xs0 - c[j]);
      acc[j] += htanh(xs1 - c[j]);
      acc[j] += htanh(xs2 - c[j]);
      acc[j] += htanh(xs3 - c[j]);
    }
  }

  // Stage per-wave partials: lds[edge][wave]
#pragma unroll
  for (int j = 0; j < 8; ++j)
    lds[(lane + 32 * j) * 8 + wave] = acc[j];
  __syncthreads();

  // Cross-wave reduction on the matrix pipe:
  // D = ones(16x4) x B(4x16) + C  => D[.,n] = sum_k B[k][n]  (column sums).
  // B layout per ISA: VGPR0 = K{0 | 2} for lanes {0-15 | 16-31}, VGPR1 = K{1 | 3}.
  const int n = lane & 15;
  const int kbase = (lane >> 4) * 2;  // 0 or 2
  v2f A;
  A[0] = 1.0f;
  A[1] = 1.0f;
#pragma unroll
  for (int gi = 0; gi < 2; ++gi) {
    const int g = wave * 2 + gi;  // 8 waves x 2 = 16 groups of 16 edges
    const float* src = &lds[(g * 16 + n) * 8];
    v2f B0, B1;
    B0[0] = src[kbase];      // waves 0..3
    B0[1] = src[kbase + 1];
    B1[0] = src[kbase + 4];  // waves 4..7
    B1[1] = src[kbase + 5];
    v8f D = {};
    D = __builtin_amdgcn_wmma_f32_16x16x4_f32(false, A, false, B0, (short)0, D,
                                              false, false);
    D = __builtin_amdgcn_wmma_f32_16x16x4_f32(false, A, false, B1, (short)0, D,
                                              false, false);
    // D VGPR0, lanes 0-15: row M=0, col N=lane -> total over all 8 waves.
    if (lane < 16) atomicAdd(&T[batch * 256 + g * 16 + lane], D[0]);
  }
}

__global__ void hist_finalize(const float* __restrict__ T,
                              float* __restrict__ out, float scale, int total) {
  const int i = blockIdx.x * blockDim.x + threadIdx.x;
  if (i >= total) return;
  const int b = i / 255;
  const int k = i % 255;
  out[i] = (T[b * 256 + k] - T[b * 256 + k + 1]) * scale;
}

extern "C" void kernel_launch(void* const* d_in, const int* in_sizes, int n_in,
                              void* d_out, int out_size, void* d_ws,
                              size_t ws_size, hipStream_t stream) {
  const float* x = (const float*)d_in[0];
  float* out = (float*)d_out;
  float* T = (float*)d_ws;  // [B, 256] edge tanh sums

  const int B = out_size / 255;        // 4
  const int N = in_sizes[0] / B;       // 262144

  hipMemsetAsync(T, 0, (size_t)B * 256 * sizeof(float), stream);

  const int blocksPerBatch = 128;
  const int pixPerBlock = N / blocksPerBatch;  // 2048 (pixPerWave = 256)

  hist_accum<<<dim3(blocksPerBatch, B), 256, 0, stream>>>(x, T, N, pixPerBlock);

  const int total = B * 255;
  hist_finalize<<<(total + 255) / 256, 256, 0, stream>>>(
      T, out, 0.5f / (float)N, total);
}